// GatedSpatialMoE2d_s_7971459301718
// MI455X (gfx1250) — compile-verified
//
#include <hip/hip_runtime.h>
#include <hip/hip_bf16.h>
#include <cstddef>

// Problem constants (from reference): N=16, E=128, H=128, W=128, K=8
#define CN 16
#define CE 128
#define CH 128
#define CW 128
#define CK 8
#define PIX 64          // pixels per workgroup (half a W-row)
#define HWSZ (CH * CW)  // 16384

typedef __attribute__((ext_vector_type(2))) float v2f;
typedef __attribute__((ext_vector_type(8))) float v8f;

// Swizzled LDS index for the experts tile: row e, pixel p.
// Rotation by 8*e keeps float4 quads contiguous (multiple of 4, no 64-wrap)
// and gives lane-halves (delta-e = 2 -> bank shift 16) disjoint bank sets,
// making the WMMA B-operand reads bank-conflict-free.
__device__ __forceinline__ int exp_idx(int e, int p) {
    return e * PIX + ((p + (e << 3)) & 63);
}

__global__ __launch_bounds__(256) void
gated_smoe2d_kernel(const float* __restrict__ experts,
                    const float* __restrict__ gate_w,
                    const float* __restrict__ gate_b,
                    float* __restrict__ out) {
    __shared__ float lds_exp[CE * PIX];   // 32 KB, swizzled [e][p]
    __shared__ float lds_log[CE * PIX];   // 32 KB, [f][p] stride 64 (phase-2 conflict-free)

    const int b      = blockIdx.x;        // 0 .. N*H*(W/64)-1 = 4095
    const int wchunk = b & 1;
    const int h      = (b >> 1) & (CH - 1);
    const int n      = b >> 8;
    const int w0     = wchunk * PIX;

    const int t    = threadIdx.x;         // 0..255 (8 waves of 32)
    const int lane = t & 31;
    const int wave = t >> 5;
    const int lrow = lane & 15;           // M / pixel-in-tile index
    const int lhi  = lane >> 4;           // 0: lanes 0-15, 1: lanes 16-31

    // ---------------- Phase 0: experts[n, :, h, w0:w0+64] -> LDS (swizzled) ----
    const size_t gbase = ((size_t)n * CE) * HWSZ + (size_t)h * CW + w0;
#pragma unroll
    for (int i = 0; i < 8; ++i) {
        const int flat = i * 256 + t;     // 0..2047
        const int e    = flat >> 4;
        const int c    = flat & 15;
        const float4 v = *reinterpret_cast<const float4*>(
            experts + gbase + (size_t)e * HWSZ + (size_t)(c * 4));
        *reinterpret_cast<float4*>(&lds_exp[exp_idx(e, c * 4)]) = v;
    }
    __syncthreads();

    // ---------------- Phase 1: gating GEMM via V_WMMA_F32_16X16X4_F32 ----------
    // Wave w owns f-tiles {2q, 2q+1} (q = w&3) and ptiles {2h, 2h+1} (h = w>>2):
    // each B fragment feeds two WMMAs (halves LDS B traffic) and the two
    // accumulator chains are independent (2x WMMA ILP).
    const int q    = wave & 3;
    const int half = wave >> 2;
    const int ft0  = 2 * q;
    const int ft1  = 2 * q + 1;

    // A layout (32-bit 16x4): lane holds M=lrow; VGPR0 = K 0/2 (lo/hi half), VGPR1 = K 1/3.
    v2f a0[32], a1[32];
    {
        const float* gw0 = gate_w + (size_t)(ft0 * 16 + lrow) * CE;
        const float* gw1 = gate_w + (size_t)(ft1 * 16 + lrow) * CE;
#pragma unroll
        for (int k = 0; k < 32; ++k) {
            const int e = 4 * k + 2 * lhi;
            a0[k].x = gw0[e];
            a0[k].y = gw0[e + 1];
            a1[k].x = gw1[e];
            a1[k].y = gw1[e + 1];
        }
    }

    for (int i = 0; i < 2; ++i) {
        const int pt = 2 * half + i;
        const int p  = pt * 16 + lrow;
        v8f acc0 = {};
        v8f acc1 = {};
#pragma unroll
        for (int k = 0; k < 32; ++k) {
            const int e   = 4 * k + 2 * lhi;   // even; pair rows e, e+1
            const int idx = exp_idx(e, p);     // row e+1 sits +64 dwords, same swizzle phase? no:
            v2f bf;
            bf.x = lds_exp[idx];
            bf.y = lds_exp[exp_idx(e + 1, p)];
            acc0 = __builtin_amdgcn_wmma_f32_16x16x4_f32(
                false, a0[k], false, bf, (short)0, acc0, false, false);
            acc1 = __builtin_amdgcn_wmma_f32_16x16x4_f32(
                false, a1[k], false, bf, (short)0, acc1, false, false);
        }
        // D layout: VGPR r -> M = r (lanes 0-15) / r+8 (lanes 16-31); N = lrow.
#pragma unroll
        for (int r = 0; r < 8; ++r) {
            const int f0 = ft0 * 16 + r + 8 * lhi;
            const int f1 = ft1 * 16 + r + 8 * lhi;
            lds_log[f0 * PIX + p] = acc0[r] + gate_b[f0];
            lds_log[f1 * PIX + p] = acc1[r] + gate_b[f1];
        }
    }
    __syncthreads();

    // ---------------- Phase 2: softmax + top-8 + gather (waves 0-1, uniform) ---
    if (t < PIX) {                        // wave-uniform: waves 0,1 fully on, 2-7 fully off
        const int p = t;
        float tv[CK];
        int   ti[CK];
#pragma unroll
        for (int j = 0; j < CK; ++j) { tv[j] = -__builtin_inff(); ti[j] = 0; }

        // Top-8 of logits (softmax is monotonic). Strict '>' keeps lax.top_k's
        // lowest-index tie-break and descending value order.
        for (int f = 0; f < CE; ++f) {
            float cv = lds_log[f * PIX + p];
            int   ci = f;
#pragma unroll
            for (int j = 0; j < CK; ++j) {
                if (cv > tv[j]) {
                    const float ftmp = tv[j]; tv[j] = cv; cv = ftmp;
                    const int   itmp = ti[j]; ti[j] = ci; ci = itmp;
                }
            }
        }

        const float mx = tv[0];           // top-1 == row max
        float denom = 0.0f;
        for (int f = 0; f < CE; ++f)
            denom += __expf(lds_log[f * PIX + p] - mx);
        const float inv = 1.0f / denom;

        float* op = out + (((size_t)n * CK) * CH + h) * CW + w0 + p;
#pragma unroll
        for (int k = 0; k < CK; ++k) {
            const float wgt = __expf(tv[k] - mx) * inv;
            op[(size_t)k * HWSZ] = lds_exp[exp_idx(ti[k], p)] * wgt;
        }
    }
}

extern "C" void kernel_launch(void* const* d_in, const int* in_sizes, int n_in,
                              void* d_out, int out_size, void* d_ws, size_t ws_size,
                              hipStream_t stream) {
    // setup_inputs order: x (unused), experts, gate_w, gate_b
    const float* experts = (const float*)d_in[1];
    const float* gate_w  = (const float*)d_in[2];
    const float* gate_b  = (const float*)d_in[3];
    float* out = (float*)d_out;

    const int blocks = CN * CH * (CW / PIX);  // 16*128*2 = 4096
    gated_smoe2d_kernel<<<blocks, 256, 0, stream>>>(experts, gate_w, gate_b, out);
}